// MultiHeadAttention_34634616275009
// MI455X (gfx1250) — compile-verified
//
#include <hip/hip_runtime.h>

typedef float v2f __attribute__((ext_vector_type(2)));
typedef float v8f __attribute__((ext_vector_type(8)));

#define Bx 2
#define Sx 2048
#define Dx 1024
#define Hx 16
#define DHx 64

__device__ __forceinline__ float rmax16(float v) {
    // reduce across the 16-lane half-wave (xor of bits 0..3 never crosses bit 4)
    for (int o = 1; o < 16; o <<= 1) v = fmaxf(v, __shfl_xor(v, o));
    return v;
}
__device__ __forceinline__ float rsum16(float v) {
    for (int o = 1; o < 16; o <<= 1) v += __shfl_xor(v, o);
    return v;
}

// ---------------------------------------------------------------------------
// Fused QKV GEMM: [4096,1024] x [1024,1024] (x3) using V_WMMA_F32_16X16X4_F32.
// One wave per 16x16 tile; x-fragment loaded once per K-step, reused 3x.
// ---------------------------------------------------------------------------
__global__ __launch_bounds__(256) void mha_qkv_kernel(
    const float* __restrict__ x,
    const float* __restrict__ W1, const float* __restrict__ b1,
    const float* __restrict__ W2, const float* __restrict__ b2,
    const float* __restrict__ W3, const float* __restrict__ b3,
    float* __restrict__ Q, float* __restrict__ K, float* __restrict__ V)
{
    const int lane = threadIdx.x & 31;
    const int wave = threadIdx.x >> 5;
    const int tile = blockIdx.x * 8 + wave;          // 16384 tiles total
    const int mt = tile >> 6;                        // / (Dx/16)
    const int nt = tile & 63;
    const int m0 = mt * 16, n0 = nt * 16;
    const int ln  = lane & 15;
    const int hi  = lane >> 4;
    const int grp = hi * 2;                          // K interleave {0,2}

    v8f aq = {}, ak = {}, av = {};
    const float* xrow = x + (size_t)(m0 + ln) * Dx;

    for (int k0 = 0; k0 < Dx; k0 += 4) {
        v2f a = *(const v2f*)(xrow + k0 + grp);      // A[M][k0+grp .. +1]
        const int kr0 = (k0 + grp) * Dx + n0 + ln;   // B[k][n], next k is +Dx
        v2f bq, bk, bv;
        bq.x = W1[kr0]; bq.y = W1[kr0 + Dx];
        bk.x = W2[kr0]; bk.y = W2[kr0 + Dx];
        bv.x = W3[kr0]; bv.y = W3[kr0 + Dx];
        aq = __builtin_amdgcn_wmma_f32_16x16x4_f32(false, a, false, bq, (short)0, aq, false, false);
        ak = __builtin_amdgcn_wmma_f32_16x16x4_f32(false, a, false, bk, (short)0, ak, false, false);
        av = __builtin_amdgcn_wmma_f32_16x16x4_f32(false, a, false, bv, (short)0, av, false, false);
    }

    const float biasq = b1[n0 + ln];
    const float biask = b2[n0 + ln];
    const float biasv = b3[n0 + ln];
    for (int r = 0; r < 8; ++r) {
        const size_t o = (size_t)(m0 + r + hi * 8) * Dx + n0 + ln;
        Q[o] = aq[r] + biasq;
        K[o] = ak[r] + biask;
        V[o] = av[r] + biasv;
    }
}

// ---------------------------------------------------------------------------
// Attention: one wave per (b, h, 16-row q-tile).
// Pass A: online softmax stats (max, sumexp) over all 2048 keys (unmasked).
// Pass B: recompute scores, attn = mask? exp(s-m)/sum : 1e-9 (mask AFTER
// softmax, per reference), write attn, accumulate ctx = attn @ V via WMMA.
// ---------------------------------------------------------------------------
__global__ __launch_bounds__(256) void mha_attn_kernel(
    const float* __restrict__ Q, const float* __restrict__ K,
    const float* __restrict__ V, const int* __restrict__ mask,
    float* __restrict__ out, float* __restrict__ attn)
{
    __shared__ float lds[8 * 256];                   // 16x16 tile per wave
    const int lane = threadIdx.x & 31;
    const int wave = threadIdx.x >> 5;
    float* wl = lds + wave * 256;

    const int tile = blockIdx.x * 8 + wave;          // 4096 tiles total
    const int bh = tile >> 7;                        // / (Sx/16)
    const int qt = tile & 127;
    const int b  = bh >> 4, h = bh & 15;
    const int q0 = qt * 16;
    const int hoff = h * DHx;
    const int ln  = lane & 15;
    const int hi  = lane >> 4;
    const int grp = hi * 2;
    const float scale = 0.03125f;                    // 1/sqrt(D) = 1/32

    const float* qbase = Q + (size_t)(b * Sx + q0 + ln) * Dx + hoff;

    float mrow[8], srow[8];
    for (int r = 0; r < 8; ++r) { mrow[r] = -INFINITY; srow[r] = 0.0f; }

    // ---- Pass A: softmax statistics over full key range ----
    for (int k0 = 0; k0 < Sx; k0 += 16) {
        const float* kbase = K + (size_t)(b * Sx + k0 + ln) * Dx + hoff;
        v8f c = {};
        for (int dh = 0; dh < DHx; dh += 4) {
            v2f a  = *(const v2f*)(qbase + dh + grp);
            v2f bb = *(const v2f*)(kbase + dh + grp);
            c = __builtin_amdgcn_wmma_f32_16x16x4_f32(false, a, false, bb, (short)0, c, false, false);
        }
        for (int r = 0; r < 8; ++r) {
            const float v  = c[r] * scale;
            const float tm = rmax16(v);
            const float nm = fmaxf(mrow[r], tm);
            const float ts = rsum16(__expf(v - nm));
            srow[r] = srow[r] * __expf(mrow[r] - nm) + ts;
            mrow[r] = nm;
        }
    }

    // ---- Pass B: attn output + ctx accumulation ----
    v8f acc0 = {}, acc1 = {}, acc2 = {}, acc3 = {};
    for (int k0 = 0; k0 < Sx; k0 += 16) {
        const float* kbase = K + (size_t)(b * Sx + k0 + ln) * Dx + hoff;
        v8f c = {};
        for (int dh = 0; dh < DHx; dh += 4) {
            v2f a  = *(const v2f*)(qbase + dh + grp);
            v2f bb = *(const v2f*)(kbase + dh + grp);
            c = __builtin_amdgcn_wmma_f32_16x16x4_f32(false, a, false, bb, (short)0, c, false, false);
        }
        __syncthreads();   // previous iteration's LDS reads complete
        for (int r = 0; r < 8; ++r) {
            const int rowM = r + hi * 8;
            const int qrow = q0 + rowM;
            const int kcol = k0 + ln;
            const float p  = __expf(c[r] * scale - mrow[r]) / srow[r];
            const float av = (mask[(size_t)qrow * Sx + kcol] == 0) ? 1e-9f : p;
            attn[((size_t)bh * Sx + qrow) * Sx + kcol] = av;
            wl[rowM * 16 + ln] = av;
        }
        __syncthreads();   // LDS tile visible; re-layout C-frag -> A-frag
        for (int kk = 0; kk < 4; ++kk) {
            v2f a2;
            a2.x = wl[ln * 16 + kk * 4 + grp];
            a2.y = wl[ln * 16 + kk * 4 + grp + 1];
            const float* vrow = V + (size_t)(b * Sx + k0 + kk * 4 + grp) * Dx + hoff + ln;
            v2f vb0, vb1, vb2, vb3;
            vb0.x = vrow[0];  vb0.y = vrow[Dx + 0];
            vb1.x = vrow[16]; vb1.y = vrow[Dx + 16];
            vb2.x = vrow[32]; vb2.y = vrow[Dx + 32];
            vb3.x = vrow[48]; vb3.y = vrow[Dx + 48];
            acc0 = __builtin_amdgcn_wmma_f32_16x16x4_f32(false, a2, false, vb0, (short)0, acc0, false, false);
            acc1 = __builtin_amdgcn_wmma_f32_16x16x4_f32(false, a2, false, vb1, (short)0, acc1, false, false);
            acc2 = __builtin_amdgcn_wmma_f32_16x16x4_f32(false, a2, false, vb2, (short)0, acc2, false, false);
            acc3 = __builtin_amdgcn_wmma_f32_16x16x4_f32(false, a2, false, vb3, (short)0, acc3, false, false);
        }
    }

    for (int r = 0; r < 8; ++r) {
        const size_t o = (size_t)(b * Sx + q0 + r + hi * 8) * Dx + hoff + ln;
        out[o]      = acc0[r];   // merged heads: out[b, s, h*DH + d]
        out[o + 16] = acc1[r];
        out[o + 32] = acc2[r];
        out[o + 48] = acc3[r];
    }
}

extern "C" void kernel_launch(void* const* d_in, const int* in_sizes, int n_in,
                              void* d_out, int out_size, void* d_ws, size_t ws_size,
                              hipStream_t stream) {
    const float* x    = (const float*)d_in[0];
    const int*   mask = (const int*)d_in[1];
    const float* W1   = (const float*)d_in[2];
    const float* b1   = (const float*)d_in[3];
    const float* W2   = (const float*)d_in[4];
    const float* b2   = (const float*)d_in[5];
    const float* W3   = (const float*)d_in[6];
    const float* b3   = (const float*)d_in[7];

    float* out  = (float*)d_out;
    float* attn = out + (size_t)Bx * Sx * Dx;        // tuple: (out, attn) flat

    const size_t mat = (size_t)Bx * Sx * Dx;         // 4,194,304 floats
    float* Q = (float*)d_ws;
    float* K = Q + mat;
    float* V = K + mat;

    // 16384 tiles (256 m-tiles x 64 n-tiles), 8 waves/block
    mha_qkv_kernel<<<2048, 256, 0, stream>>>(x, W1, b1, W2, b2, W3, b3, Q, K, V);
    // 4096 tiles (2 b x 16 h x 128 q-tiles), 8 waves/block
    mha_attn_kernel<<<512, 256, 0, stream>>>(Q, K, V, mask, out, attn);
}